// OLA_87565793230895
// MI455X (gfx1250) — compile-verified
//
#include <hip/hip_runtime.h>
#include <cstdint>
#include <cstddef>

// -----------------------------------------------------------------------------
// Overlap-add (OLA), B=16, F=4000, L=512, hop=128, trim=384 each side.
// out[b][t] = sum_{k=0..3} x[b, (t+384)>>7 - k, ((t+384)&127) + 128k]
// Memory-bound: ~164 MB total traffic -> ~7us at 23.3 TB/s on MI455X.
// CDNA5 path: stage each 128-output chunk's 2KB diagonal band into LDS with
// GLOBAL_LOAD_ASYNC_TO_LDS_B128 (ASYNCcnt), s_wait_asynccnt, barrier, reduce.
// Addressing kept as uniform-SGPR base + 32-bit lane offset to minimize VALU.
// -----------------------------------------------------------------------------

namespace {
constexpr int kFft     = 512;
constexpr int kHop     = 128;
constexpr int kOvl     = 384;                                   // 3 * hop
constexpr int kBatch   = 16;
constexpr int kFrames  = 4000;
constexpr int kOutLen  = (kFrames - 1) * kHop + kFft - 2 * kOvl; // 511616
constexpr int kChunks  = kOutLen / 128;                          // 3997 (exact)
constexpr int kCpb     = 8;                                      // chunks / block
constexpr int kBlock   = 256;                                    // 8 waves (wave32)
constexpr int kGridX   = (kChunks + kCpb - 1) / kCpb;            // 500
}

typedef int v4i __attribute__((ext_vector_type(4)));
typedef __attribute__((address_space(1))) v4i* glb_v4i_p;
typedef __attribute__((address_space(3))) v4i* lds_v4i_p;
typedef __attribute__((address_space(3))) void* lds_void_p;

#if defined(__has_builtin)
#if __has_builtin(__builtin_amdgcn_global_load_async_to_lds_b128)
#define OLA_ASYNC_BUILTIN 1
#endif
#if __has_builtin(__builtin_amdgcn_s_wait_asynccnt)
#define OLA_WAIT_BUILTIN 1
#endif
#endif

__device__ __forceinline__ void ola_async_ld_b128(const float* g, float* l) {
#if defined(OLA_ASYNC_BUILTIN)
  // (global src v4i*, lds dst v4i*, imm offset, imm cpol)
  __builtin_amdgcn_global_load_async_to_lds_b128(
      (glb_v4i_p)g, (lds_v4i_p)l, 0, 0);
#else
  unsigned lds_off = (unsigned)(uintptr_t)(lds_void_p)l;    // 32-bit LDS byte addr
  unsigned long long ga = (unsigned long long)(uintptr_t)g; // 64-bit global addr
  asm volatile("global_load_async_to_lds_b128 %0, %1, off"
               :
               : "v"(lds_off), "v"(ga)
               : "memory");
#endif
}

__device__ __forceinline__ void ola_wait_async() {
#if defined(OLA_WAIT_BUILTIN)
  __builtin_amdgcn_s_wait_asynccnt(0);
#else
  asm volatile("s_wait_asynccnt 0" ::: "memory");
#endif
}

__global__ __launch_bounds__(kBlock) void
OLA_87565793230895_kernel(const float* __restrict__ x, float* __restrict__ out) {
  __shared__ float smem[kCpb * 512];          // 8 chunks * 4 rows * 128 floats = 16 KB

  const int tid = threadIdx.x;
  const int b   = blockIdx.y;
  const int q0  = blockIdx.x * kCpb;

  // Uniform (scalar) bases; per-lane address = base + small 32-bit offset.
  const float* gbase = x
      + (size_t)b * (size_t)(kFrames * kFft)   // batch row
      + (size_t)q0 * kFft + kOvl;              // &x[b, q0, 384]
  float* obase = out + (size_t)b * kOutLen + (size_t)q0 * 128;

  // --- Stage: per chunk q = q0+cl, the 4 needed 128-float segments form a
  //     diagonal band reachable from one base with constant stride:
  //     segment y (=0..3) at gbase + cl*512 + y*384  (row q+y, cols 384-128y..).
  //     128 b128 transfers per chunk, fully coalesced 512B runs.
#pragma unroll
  for (int r = 0; r < (kCpb * 128) / kBlock; ++r) {   // 4 iterations
    const int it = tid + r * kBlock;                  // transfer id, 0..1023
    const int cl = it >> 7;                           // chunk-local 0..7
    const int w  = it & 127;                          // transfer within chunk
    if (q0 + cl < kChunks) {
      const int y  = w >> 5;                          // band row 0..3
      const int xo = w & 31;                          // float4 within segment
      const int goff = cl * kFft + y * 384 + xo * 4;  // 32-bit lane offset (floats)
      ola_async_ld_b128(gbase + goff, &smem[it * 4]);
    }
  }
  ola_wait_async();      // own wave's async transfers landed in LDS
  __syncthreads();       // make all waves' staged data visible

  // --- Reduce: out[b][128*(q0+cl) + m] = sum_y smem[512*cl + 128*y + m]
  const int o  = tid << 2;        // 0..1020, one float4 per thread; o = cl*128 + m
  const int cl = o >> 7;
  const int m  = o & 127;
  if (q0 + cl < kChunks) {
    const float* s = &smem[cl * 512 + m];
    const float4 s0 = *reinterpret_cast<const float4*>(s + 0);
    const float4 s1 = *reinterpret_cast<const float4*>(s + 128);
    const float4 s2 = *reinterpret_cast<const float4*>(s + 256);
    const float4 s3 = *reinterpret_cast<const float4*>(s + 384);
    float4 rv;
    rv.x = (s0.x + s1.x) + (s2.x + s3.x);
    rv.y = (s0.y + s1.y) + (s2.y + s3.y);
    rv.z = (s0.z + s1.z) + (s2.z + s3.z);
    rv.w = (s0.w + s1.w) + (s2.w + s3.w);
    *reinterpret_cast<float4*>(obase + o) = rv;       // coalesced, write-once
  }
}

extern "C" void kernel_launch(void* const* d_in, const int* in_sizes, int n_in,
                              void* d_out, int out_size, void* d_ws, size_t ws_size,
                              hipStream_t stream) {
  (void)in_sizes; (void)n_in; (void)d_ws; (void)ws_size; (void)out_size;
  const float* x = (const float*)d_in[0];  // [16, 4000, 512] f32
  float* out = (float*)d_out;              // [16, 511616] f32
  dim3 grid(kGridX, kBatch);
  OLA_87565793230895_kernel<<<grid, dim3(kBlock), 0, stream>>>(x, out);
}